// SSMFwd_15676630631258
// MI455X (gfx1250) — compile-verified
//
#include <hip/hip_runtime.h>

typedef __attribute__((ext_vector_type(2))) float v2f;
typedef __attribute__((ext_vector_type(4))) float v4f;
typedef __attribute__((ext_vector_type(8))) float v8f;

#define DDIM 256
#define BS   16
#define TLEN 4096
#define CHUNK 64
#define NCH  (TLEN / CHUNK)          // 64 chunks
#define MATELEMS (DDIM * DDIM)       // 65536
#define HPAD 260                     // padded LDS row stride (bank-conflict-free)

__device__ __forceinline__ v8f wmma4(v2f a, v2f b, v8f c) {
  // D = A(16x4,f32) * B(4x16,f32) + C(16x16,f32)
  return __builtin_amdgcn_wmma_f32_16x16x4_f32(false, a, false, b, (short)0, c,
                                               false, false);
}

__device__ __forceinline__ v8f vzero8() {
  v8f z = {0.f, 0.f, 0.f, 0.f, 0.f, 0.f, 0.f, 0.f};
  return z;
}

// ---------------------------------------------------------------- powers ----
__global__ void copy_mat(const float* __restrict__ src, float* __restrict__ dst) {
  int i = blockIdx.x * blockDim.x + threadIdx.x;
  if (i < MATELEMS) dst[i] = src[i];
}

// powers[k] lives at pw + (k-1)*MATELEMS.  Computes A_{s+j} = A_s * A_j,
// j = blockIdx.y+1 (1..s).  8 waves/WG, one 16x16 output tile per wave.
__global__ void power_step(float* __restrict__ pw, int s) {
  const int j = blockIdx.y + 1;
  const float* As = pw + (size_t)(s - 1) * MATELEMS;
  const float* Aj = pw + (size_t)(j - 1) * MATELEMS;
  float*       Dm = pw + (size_t)(s + j - 1) * MATELEMS;

  const int lane = threadIdx.x & 31;
  const int wave = threadIdx.x >> 5;        // 0..7
  const int tile = blockIdx.x * 8 + wave;   // 0..255
  const int mt = tile >> 4, nt = tile & 15;
  const int col = lane & 15;
  const int kk  = (lane >> 4) * 2;

  v8f acc = vzero8();
  const float* arow = As + (mt * 16 + col) * DDIM;
  for (int e = 0; e < DDIM; e += 4) {
    v2f a = *(const v2f*)(arow + e + kk);
    v2f b;
    b.x = Aj[(e + kk    ) * DDIM + nt * 16 + col];
    b.y = Aj[(e + kk + 1) * DDIM + nt * 16 + col];
    acc = wmma4(a, b, acc);
  }
  const int hi = (lane >> 4) * 8;
#pragma unroll
  for (int v = 0; v < 8; ++v)
    Dm[(mt * 16 + v + hi) * DDIM + nt * 16 + col] = acc[v];
}

// ---------------------------------------------------------------- Bx GEMM ---
// Bx[r,n] = sum_e X[r,e] * Bm[n,e]  (Bx = X @ Bm^T).  8 waves/WG; each wave
// owns a 16x64 strip (A fragment reused across 4 WMMAs per k-step).
__global__ void bx_gemm(const float* __restrict__ X, const float* __restrict__ Bm,
                        float* __restrict__ Bx) {
  const int lane = threadIdx.x & 31;
  const int wave = threadIdx.x >> 5;      // 0..7
  const int r0 = blockIdx.x * 128 + wave * 16;
  const int n0 = blockIdx.y * 64;
  const int col = lane & 15;
  const int kk  = (lane >> 4) * 2;

  v8f acc0 = vzero8(), acc1 = vzero8(), acc2 = vzero8(), acc3 = vzero8();
  const float* xrow = X  + (r0 + col) * DDIM;
  const float* b0   = Bm + (n0 +  0 + col) * DDIM;
  const float* b1   = Bm + (n0 + 16 + col) * DDIM;
  const float* b2   = Bm + (n0 + 32 + col) * DDIM;
  const float* b3   = Bm + (n0 + 48 + col) * DDIM;
  for (int e = 0; e < DDIM; e += 4) {
    v2f a = *(const v2f*)(xrow + e + kk);
    acc0 = wmma4(a, *(const v2f*)(b0 + e + kk), acc0);
    acc1 = wmma4(a, *(const v2f*)(b1 + e + kk), acc1);
    acc2 = wmma4(a, *(const v2f*)(b2 + e + kk), acc2);
    acc3 = wmma4(a, *(const v2f*)(b3 + e + kk), acc3);
  }
  const int hi = (lane >> 4) * 8;
#pragma unroll
  for (int v = 0; v < 8; ++v) {
    float* orow = Bx + (r0 + v + hi) * DDIM + col;
    orow[n0 +  0] = acc0[v];
    orow[n0 + 16] = acc1[v];
    orow[n0 + 32] = acc2[v];
    orow[n0 + 48] = acc3[v];
  }
}

// ------------------------------------------------------------- chunk scan ---
// Per chunk (blockIdx.x): u_i = Lam @ u_{i-1} + Bx_t for i=0..CHUNK-1 with
// u_{-1}=0, all 16 batches as the M dim of a 16x256x256 WMMA GEMM per step.
// Lam (256KB) is staged into the 320KB WGP LDS via CDNA5 async global->LDS
// b128 copies; h ping-pongs through LDS.  16 waves, wave w owns cols [16w..).
// Dynamic LDS layout: [0, 256*HPAD) = Lam (padded rows); then 2*16*HPAD for h.
__global__ void __launch_bounds__(512) chunk_scan(const float* __restrict__ Bx,
                                                  const float* __restrict__ Lam,
                                                  float* __restrict__ U) {
  extern __shared__ float smem[];
  float* lam_s = smem;                       // 256 x HPAD
  float* hb    = smem + 256 * HPAD;          // 2 x BS x HPAD

  const int lane = threadIdx.x & 31;
  const int wave = threadIdx.x >> 5;     // 0..15
  const int chunk = blockIdx.x;
  const int n0 = wave * 16;
  const int col = lane & 15;
  const int kk  = (lane >> 4) * 2;
  const int hi  = (lane >> 4) * 8;

  // ---- stage Lam into LDS with async global->LDS copies (ASYNCcnt path).
  // Dynamic LDS starts at group-segment offset 0, so LDS byte offsets are
  // computed numerically.  Each b128 moves 4 floats of one (padded) row.
  {
    const unsigned long long gbase = (unsigned long long)(uintptr_t)Lam;
    for (int g = threadIdx.x; g < MATELEMS / 4; g += (int)blockDim.x) {
      const int row = g >> 6;              // 64 x b128 per 256-float row
      const int e   = (g & 63) * 4;
      unsigned ldsoff = (unsigned)((row * HPAD + e) * 4);
      unsigned voff   = (unsigned)(g * 16);
      asm volatile("global_load_async_to_lds_b128 %0, %1, %2"
                   :: "v"(ldsoff), "v"(voff), "s"(gbase)
                   : "memory");
    }
    asm volatile("s_wait_asynccnt 0x0" ::: "memory");
  }

  // zero h_{-1} (buffer 0)
  for (int i = threadIdx.x; i < BS * DDIM; i += (int)blockDim.x)
    hb[(i >> 8) * HPAD + (i & 255)] = 0.f;
  __syncthreads();

  const float* lrow = lam_s + (n0 + col) * HPAD;   // row n of Lam: B[k,n]=Lam[n,k]
  for (int i = 0; i < CHUNK; ++i) {
    const int t = chunk * CHUNK + i;
    const float* hsrc = hb + (i & 1) * (BS * HPAD);
    v8f acc = vzero8();
    for (int e = 0; e < DDIM; e += 4) {
      v2f a = *(const v2f*)(hsrc + col * HPAD + e + kk);  // A[m=batch,k]
      v2f b = *(const v2f*)(lrow + e + kk);
      acc = wmma4(a, b, acc);
    }
    float* hdst = hb + ((i + 1) & 1) * (BS * HPAD);
#pragma unroll
    for (int v = 0; v < 8; ++v) {
      const int bat = v + hi;
      float h = acc[v] + Bx[(bat * TLEN + t) * DDIM + n0 + col];
      hdst[bat * HPAD + n0 + col] = h;
      U[(bat * TLEN + t) * DDIM + n0 + col] = h;
    }
    __syncthreads();
  }
}

// ------------------------------------------------------------- carry scan ---
// carry_k = Lam^CHUNK @ carry_{k-1} + u_last(k).  One WG per batch; thread n
// owns output dim n; true mat-vec -> VALU fmacs.  P staged into LDS once.
__global__ void carry_scan(const float* __restrict__ U, const float* __restrict__ P,
                           float* __restrict__ carries) {
  extern __shared__ float smem[];
  float* p_s = smem;                  // 256 x HPAD
  float* hc  = smem + 256 * HPAD;     // 256

  const int b = blockIdx.x;
  const int n = threadIdx.x;

  for (int g = threadIdx.x; g < MATELEMS / 4; g += (int)blockDim.x) {
    const int row = g >> 6;
    const int e   = (g & 63) * 4;
    *(v4f*)(p_s + row * HPAD + e) = *(const v4f*)(P + g * 4);
  }
  hc[n] = 0.f;
  __syncthreads();

  const float* prow = p_s + n * HPAD;
  for (int k = 0; k < NCH; ++k) {
    float s = U[(b * TLEN + (k * CHUNK + CHUNK - 1)) * DDIM + n];
    for (int e = 0; e < DDIM; e += 4) {
      v4f p = *(const v4f*)(prow + e);
      v4f h = *(const v4f*)(hc + e);
      s += p.x * h.x + p.y * h.y + p.z * h.z + p.w * h.w;
    }
    __syncthreads();
    hc[n] = s;
    carries[(k * BS + b) * DDIM + n] = s;
    __syncthreads();
  }
}

// ------------------------------------------------------------- correction ---
// out[b, k*CHUNK+i, :] += Lam^{i+1} @ carry_{k-1}[b].  One WG per (i, k);
// accumulator seeded from out so the add is fused into the WMMA C operand.
// Powers are read exactly once per WG -> keep on the (192MB) L2 path.
__global__ void __launch_bounds__(512) correct(float* __restrict__ out,
                                               const float* __restrict__ pw,
                                               const float* __restrict__ carries) {
  const int i = blockIdx.x;            // 0..CHUNK-1
  const int k = blockIdx.y + 1;        // chunk 1..NCH-1
  const int lane = threadIdx.x & 31;
  const int wave = threadIdx.x >> 5;   // 0..15
  const int n0 = wave * 16;
  const int col = lane & 15;
  const int kk  = (lane >> 4) * 2;
  const int hi  = (lane >> 4) * 8;
  const int t = k * CHUNK + i;

  const float* carry = carries + (size_t)(k - 1) * BS * DDIM;
  const float* Amat  = pw + (size_t)i * MATELEMS;       // Lam^{i+1}
  const float* arow  = carry + col * DDIM;              // A[m=batch,k=e]
  const float* brow  = Amat + (n0 + col) * DDIM;        // B[k,n]=Lam^{i+1}[n,k]

  v8f acc;
#pragma unroll
  for (int v = 0; v < 8; ++v)
    acc[v] = out[((v + hi) * TLEN + t) * DDIM + n0 + col];
  for (int e = 0; e < DDIM; e += 4) {
    v2f a = *(const v2f*)(arow + e + kk);
    v2f b = *(const v2f*)(brow + e + kk);
    acc = wmma4(a, b, acc);
  }
#pragma unroll
  for (int v = 0; v < 8; ++v)
    out[((v + hi) * TLEN + t) * DDIM + n0 + col] = acc[v];
}

// ------------------------------------------------------------------ launch --
extern "C" void kernel_launch(void* const* d_in, const int* in_sizes, int n_in,
                              void* d_out, int out_size, void* d_ws, size_t ws_size,
                              hipStream_t stream) {
  (void)in_sizes; (void)n_in; (void)out_size; (void)ws_size;
  const float* x   = (const float*)d_in[0];   // [16,4096,256]
  const float* Bm  = (const float*)d_in[1];   // [256,256]
  const float* Lam = (const float*)d_in[2];   // [256,256]
  float* out = (float*)d_out;                 // [16,4096,256], also U buffer

  float* Bx      = (float*)d_ws;                         // 16.78M floats
  float* powers  = Bx + (size_t)BS * TLEN * DDIM;        // 64 * 65536 floats
  float* carries = powers + (size_t)CHUNK * MATELEMS;    // NCH*BS*DDIM floats

  // Lam^1..Lam^64 by log-doubling.
  copy_mat<<<MATELEMS / 256, 256, 0, stream>>>(Lam, powers);
  for (int s = 1; s < CHUNK; s <<= 1)
    power_step<<<dim3(32, s), 256, 0, stream>>>(powers, s);

  // Bx = X @ B^T
  bx_gemm<<<dim3((BS * TLEN) / 128, DDIM / 64), 256, 0, stream>>>(x, Bm, Bx);

  // chunk-local scans: Lam lives in the 320KB WGP LDS (256*HPAD + 2*16*HPAD)
  const size_t lds_scan = (size_t)(256 * HPAD + 2 * BS * HPAD) * sizeof(float);
  chunk_scan<<<NCH, 512, lds_scan, stream>>>(Bx, Lam, out);

  // serial carry scan across chunk boundaries (P = Lam^CHUNK in LDS)
  const size_t lds_carry = (size_t)(256 * HPAD + DDIM) * sizeof(float);
  carry_scan<<<BS, DDIM, lds_carry, stream>>>(
      out, powers + (size_t)(CHUNK - 1) * MATELEMS, carries);

  // broadcast carries back into every timestep
  correct<<<dim3(CHUNK, NCH - 1), 512, 0, stream>>>(out, powers, carries);
}